// concatPGCN_11845519802990
// MI455X (gfx1250) — compile-verified
//
#include <hip/hip_runtime.h>
#include <math.h>

// ---------------------------------------------------------------------------
// PGCN forward on MI455X (gfx1250), fp32 WMMA (V_WMMA_F32_16X16X4_F32).
// Register-blocked: one wave computes a 16x64 output strip (4 accumulators),
// reusing each A fragment across 4 WMMAs. global_prefetch feeds the k-stream.
// ---------------------------------------------------------------------------

typedef __attribute__((ext_vector_type(2))) float v2f;
typedef __attribute__((ext_vector_type(8))) float v8f;

#define NN     4096
#define DX     2048   // X1D + X2D
#define X1D_   1536
#define X2D_   512
#define METAD  8
#define HG     256
#define THRESH_ 1.6f
#define BNEPS  1e-5f
#define NT     4      // 16-wide N tiles per wave (16x64 strip)

// ---------------- block reduction helper ----------------
__device__ __forceinline__ float block_reduce_sum(float v, float* sh) {
    int tid = threadIdx.x;
    sh[tid] = v;
    __syncthreads();
    for (int s = blockDim.x >> 1; s > 0; s >>= 1) {
        if (tid < s) sh[tid] += sh[tid + s];
        __syncthreads();
    }
    float r = sh[0];
    __syncthreads();
    return r;
}

// ---------------- WMMA f32 16x16x4 multi-tile core ----------------
// A: row-major M x K (lda). B: row-major K x N (ldb)  [BT=false]
//                           or row-major N x K (ldb), used transposed [BT=true]
// One wave accumulates NT 16x16 tiles at (m0, n0..n0+NT*16).
template <bool BT>
__device__ __forceinline__ void wmma_multi(const float* __restrict__ A,
                                           const float* __restrict__ B,
                                           int lda, int ldb,
                                           int m0, int n0, int K, v8f* acc) {
    const int lane = threadIdx.x & 31;
    const int half = lane >> 4;   // 0: K={0,1}, 1: K={2,3} per 4-step
    const int r    = lane & 15;   // A row (m) / B col (n) within tile
    const float* arow = A + (size_t)(m0 + r) * lda;
    const float* brow[NT];
    if (BT) {
#pragma unroll
        for (int t = 0; t < NT; ++t)
            brow[t] = B + (size_t)(n0 + t * 16 + r) * ldb;
    }
    if ((K & 63) == 0) {
        for (int kbo = 0; kbo < K; kbo += 64) {
            __builtin_prefetch(arow + kbo + 64, 0, 1);   // global_prefetch_b8
#pragma unroll
            for (int ku = 0; ku < 64; ku += 4) {
                const int k0 = kbo + ku + 2 * half;
                v2f a;
                a.x = arow[k0];
                a.y = arow[k0 + 1];
#pragma unroll
                for (int t = 0; t < NT; ++t) {
                    v2f b;
                    if (BT) {
                        b.x = brow[t][k0];
                        b.y = brow[t][k0 + 1];
                    } else {
                        b.x = B[(size_t)k0 * ldb + n0 + t * 16 + r];
                        b.y = B[(size_t)(k0 + 1) * ldb + n0 + t * 16 + r];
                    }
                    acc[t] = __builtin_amdgcn_wmma_f32_16x16x4_f32(
                        false, a, false, b, (short)0, acc[t], false, false);
                }
            }
        }
    } else {  // small-K fallback (meta distance, K=8)
        for (int kb = 0; kb < K; kb += 4) {
            const int k0 = kb + 2 * half;
            v2f a;
            a.x = arow[k0];
            a.y = arow[k0 + 1];
#pragma unroll
            for (int t = 0; t < NT; ++t) {
                v2f b;
                if (BT) {
                    b.x = brow[t][k0];
                    b.y = brow[t][k0 + 1];
                } else {
                    b.x = B[(size_t)k0 * ldb + n0 + t * 16 + r];
                    b.y = B[(size_t)(k0 + 1) * ldb + n0 + t * 16 + r];
                }
                acc[t] = __builtin_amdgcn_wmma_f32_16x16x4_f32(
                    false, a, false, b, (short)0, acc[t], false, false);
            }
        }
    }
}

// C/D 16x16 f32 layout: lane col n = lane&15, row m = vgpr + 8*(lane>>4)
#define TILE_STORE_MT(expr)                                                   \
    do {                                                                      \
        const int lane = threadIdx.x & 31;                                    \
        const int half = lane >> 4;                                           \
        const int nl   = lane & 15;                                           \
        _Pragma("unroll")                                                     \
        for (int t = 0; t < NT; ++t) {                                        \
            _Pragma("unroll")                                                 \
            for (int v = 0; v < 8; ++v) {                                     \
                const int m = v + 8 * half;                                   \
                const int n = t * 16 + nl;                                    \
                const size_t idx = (size_t)(m0 + m) * ldc + (n0 + n);         \
                (expr);                                                       \
            }                                                                 \
        }                                                                     \
    } while (0)

// ---------------- kernels ----------------

__global__ void k_zero_doubles(double* p, int n) {
    int i = blockIdx.x * blockDim.x + threadIdx.x;
    if (i < n) p[i] = 0.0;
}

// Xcat = concat(x1,x2); Xn = centered+normalized rows of Xcat
__global__ void k_rownorm_cat(const float* __restrict__ x1,
                              const float* __restrict__ x2,
                              float* __restrict__ Xcat,
                              float* __restrict__ Xn) {
    __shared__ float sh[256];
    const int row = blockIdx.x, tid = threadIdx.x;
    float s = 0.f;
    for (int j = tid; j < DX; j += 256) {
        float v = (j < X1D_) ? x1[(size_t)row * X1D_ + j]
                             : x2[(size_t)row * X2D_ + (j - X1D_)];
        Xcat[(size_t)row * DX + j] = v;
        s += v;
    }
    const float mean = block_reduce_sum(s, sh) * (1.0f / (float)DX);
    float q = 0.f;
    for (int j = tid; j < DX; j += 256) {
        float c = Xcat[(size_t)row * DX + j] - mean;
        q += c * c;
    }
    const float nrm2 = block_reduce_sum(q, sh);
    const float rn = (nrm2 > 0.f) ? rsqrtf(nrm2) : 0.f;
    for (int j = tid; j < DX; j += 256)
        Xn[(size_t)row * DX + j] = (Xcat[(size_t)row * DX + j] - mean) * rn;
}

// Xn3 = centered+normalized rows of x3 (8-wide rows, 1 thread per row)
__global__ void k_rownorm_meta(const float* __restrict__ x3,
                               float* __restrict__ Xn3) {
    const int row = blockIdx.x * blockDim.x + threadIdx.x;
    if (row >= NN) return;
    float m = 0.f;
#pragma unroll
    for (int j = 0; j < METAD; ++j) m += x3[(size_t)row * METAD + j];
    m *= (1.0f / METAD);
    float q = 0.f;
#pragma unroll
    for (int j = 0; j < METAD; ++j) {
        float c = x3[(size_t)row * METAD + j] - m;
        q += c * c;
    }
    const float rn = (q > 0.f) ? rsqrtf(q) : 0.f;
#pragma unroll
    for (int j = 0; j < METAD; ++j)
        Xn3[(size_t)row * METAD + j] = (x3[(size_t)row * METAD + j] - m) * rn;
}

// D = 1 - A*A^T (correlation distance), WMMA NT path. 8 waves/block,
// each wave: 16 rows x 64 cols.
__global__ void k_dist_nt(const float* __restrict__ A, int K, int lda,
                          float* __restrict__ D) {
    const int wave = threadIdx.x >> 5;
    const int m0 = (blockIdx.y * 8 + wave) * 16;
    const int n0 = blockIdx.x * (16 * NT);
    const int ldc = NN;
    v8f acc[NT] = {};
    wmma_multi<true>(A, A, lda, lda, m0, n0, K, acc);
    TILE_STORE_MT(D[idx] = 1.0f - acc[t][v]);
}

// fp64 grid-stride sum reduction -> atomicAdd(double)
__global__ void k_sum_f64(const float* __restrict__ X, long long n,
                          double* __restrict__ acc) {
    __shared__ double sh[256];
    const int tid = threadIdx.x;
    double s = 0.0;
    for (long long i = (long long)blockIdx.x * blockDim.x + tid; i < n;
         i += (long long)gridDim.x * blockDim.x)
        s += (double)X[i];
    sh[tid] = s;
    __syncthreads();
    for (int st = 128; st > 0; st >>= 1) {
        if (tid < st) sh[tid] += sh[tid + st];
        __syncthreads();
    }
    if (tid == 0) atomicAdd(acc, sh[0]);
}

// adj = exp(-D1^2/(2 s1^2)) + exp(-D2^2/(2 s2^2)); threshold >1.6; zero diag.
// Written in-place over D1; row degree -> deg[row].
__global__ void k_adj_deg(float* __restrict__ D1, const float* __restrict__ D2,
                          const double* __restrict__ sums,
                          float* __restrict__ deg) {
    __shared__ float sh[256];
    const int row = blockIdx.x, tid = threadIdx.x;
    const float s1 = (float)(sums[0] / ((double)NN * (double)NN));
    const float s2 = (float)(sums[1] / ((double)NN * (double)NN));
    const float c1 = -1.0f / (2.0f * s1 * s1);
    const float c2 = -1.0f / (2.0f * s2 * s2);
    float s = 0.f;
    for (int col = tid; col < NN; col += 256) {
        const size_t idx = (size_t)row * NN + col;
        const float d1 = D1[idx], d2 = D2[idx];
        float a = expf(d1 * d1 * c1) + expf(d2 * d2 * c2);
        a = (a > THRESH_) ? a : 0.f;
        if (col == row) a = 0.f;
        D1[idx] = a;
        s += a;
    }
    const float d = block_reduce_sum(s, sh);
    if (tid == 0) deg[row] = d;
}

__global__ void k_dinv(const float* __restrict__ deg, float* __restrict__ dinv) {
    const int i = blockIdx.x * blockDim.x + threadIdx.x;
    if (i < NN) dinv[i] = (deg[i] > 0.f) ? rsqrtf(deg[i]) : 0.f;
}

// Lhat = -(dinv_i * adj_ij * dinv_j), in-place
__global__ void k_lhat(float* __restrict__ A, const float* __restrict__ dinv) {
    const long long total = (long long)NN * NN;
    for (long long idx = (long long)blockIdx.x * blockDim.x + threadIdx.x;
         idx < total; idx += (long long)gridDim.x * blockDim.x) {
        const int i = (int)(idx >> 12);   // /4096
        const int j = (int)(idx & 4095);
        A[idx] = -(dinv[i] * A[idx] * dinv[j]);
    }
}

// C = alpha * A@B + beta * Dm   (A: M x K row-major lda=K; B: K x N ldb=N)
__global__ void k_gemm_nn(const float* __restrict__ A, const float* __restrict__ B,
                          const float* __restrict__ Dm, float* __restrict__ C,
                          int Ncols, int K, float alpha, float beta) {
    const int wave = threadIdx.x >> 5;
    const int m0 = (blockIdx.y * 8 + wave) * 16;
    const int n0 = blockIdx.x * (16 * NT);
    const int ldc = Ncols;
    v8f acc[NT] = {};
    wmma_multi<false>(A, B, K, Ncols, m0, n0, K, acc);
    if (beta != 0.f) {
        TILE_STORE_MT(C[idx] = alpha * acc[t][v] + beta * Dm[idx]);
    } else {
        TILE_STORE_MT(C[idx] = alpha * acc[t][v]);
    }
}

// C = relu(A0@B0 + A1@B1 + A2@B2), all Ai: M x K (lda=K), Bi: K x N (ldb=N)
__global__ void k_gemm3_relu(const float* __restrict__ A0, const float* __restrict__ A1,
                             const float* __restrict__ A2, const float* __restrict__ B0,
                             const float* __restrict__ B1, const float* __restrict__ B2,
                             float* __restrict__ C, int Ncols, int K) {
    const int wave = threadIdx.x >> 5;
    const int m0 = (blockIdx.y * 8 + wave) * 16;
    const int n0 = blockIdx.x * (16 * NT);
    const int ldc = Ncols;
    const int lane = threadIdx.x & 31;
    const int half = lane >> 4;
    const int r = lane & 15;
    v8f acc[NT] = {};
    const float* ar[3] = { A0 + (size_t)(m0 + r) * K,
                           A1 + (size_t)(m0 + r) * K,
                           A2 + (size_t)(m0 + r) * K };
    const float* Bs[3] = { B0, B1, B2 };
    for (int kbo = 0; kbo < K; kbo += 64) {
        __builtin_prefetch(ar[0] + kbo + 64, 0, 1);
        __builtin_prefetch(ar[1] + kbo + 64, 0, 1);
        __builtin_prefetch(ar[2] + kbo + 64, 0, 1);
#pragma unroll 4
        for (int ku = 0; ku < 64; ku += 4) {
            const int k0 = kbo + ku + 2 * half;
#pragma unroll
            for (int sIdx = 0; sIdx < 3; ++sIdx) {
                v2f a;
                a.x = ar[sIdx][k0];
                a.y = ar[sIdx][k0 + 1];
#pragma unroll
                for (int t = 0; t < NT; ++t) {
                    v2f b;
                    b.x = Bs[sIdx][(size_t)k0 * Ncols + n0 + t * 16 + r];
                    b.y = Bs[sIdx][(size_t)(k0 + 1) * Ncols + n0 + t * 16 + r];
                    acc[t] = __builtin_amdgcn_wmma_f32_16x16x4_f32(
                        false, a, false, b, (short)0, acc[t], false, false);
                }
            }
        }
    }
    TILE_STORE_MT(C[idx] = fmaxf(acc[t][v], 0.f));
}

// C = relu(A@B + bias)
__global__ void k_gemm_bias_relu(const float* __restrict__ A, const float* __restrict__ B,
                                 const float* __restrict__ bias, float* __restrict__ C,
                                 int Ncols, int K) {
    const int wave = threadIdx.x >> 5;
    const int m0 = (blockIdx.y * 8 + wave) * 16;
    const int n0 = blockIdx.x * (16 * NT);
    const int ldc = Ncols;
    v8f acc[NT] = {};
    wmma_multi<false>(A, B, K, Ncols, m0, n0, K, acc);
    TILE_STORE_MT(C[idx] = fmaxf(acc[t][v] + bias[n0 + n], 0.f));
}

// per-column sum and sum-of-squares over M rows (BatchNorm stats)
__global__ void k_colstats(const float* __restrict__ Z, float* __restrict__ musum,
                           float* __restrict__ sqsum, int M, int Ncols) {
    __shared__ float sh[256];
    const int c = blockIdx.x, tid = threadIdx.x;
    float s = 0.f, q = 0.f;
    for (int rr = tid; rr < M; rr += 256) {
        const float v = Z[(size_t)rr * Ncols + c];
        s += v;
        q += v * v;
    }
    const float ts = block_reduce_sum(s, sh);
    const float tq = block_reduce_sum(q, sh);
    if (tid == 0) { musum[c] = ts; sqsum[c] = tq; }
}

// out = BN(Z) @ W2 + b2   (N=2 output cols, tiny: plain VALU)
__global__ void k_final(const float* __restrict__ Z, const float* __restrict__ musum,
                        const float* __restrict__ sqsum, const float* __restrict__ gamma,
                        const float* __restrict__ beta, const float* __restrict__ W2,
                        const float* __restrict__ b2, float* __restrict__ out, int M) {
    const int r = blockIdx.x * blockDim.x + threadIdx.x;
    if (r >= M) return;
    float a0 = b2[0], a1 = b2[1];
    const float rM = 1.0f / (float)M;
    for (int j = 0; j < HG; ++j) {
        const float mu = musum[j] * rM;
        const float var = sqsum[j] * rM - mu * mu;
        const float zn = (Z[(size_t)r * HG + j] - mu) * rsqrtf(var + BNEPS) * gamma[j] + beta[j];
        a0 += zn * W2[j * 2 + 0];
        a1 += zn * W2[j * 2 + 1];
    }
    out[r * 2 + 0] = a0;
    out[r * 2 + 1] = a1;
}

// ---------------------------------------------------------------------------
extern "C" void kernel_launch(void* const* d_in, const int* in_sizes, int n_in,
                              void* d_out, int out_size, void* d_ws, size_t ws_size,
                              hipStream_t stream) {
    const float* x1     = (const float*)d_in[0];
    const float* x2     = (const float*)d_in[1];
    const float* x3     = (const float*)d_in[2];
    const float* gc_w0  = (const float*)d_in[3];   // [3, 2048, 256]
    const float* gc_w1  = (const float*)d_in[4];   // [3, 256, 256]
    const float* cls_w1 = (const float*)d_in[5];   // [256, 256]
    const float* cls_b1 = (const float*)d_in[6];
    const float* bn_g   = (const float*)d_in[7];
    const float* bn_b   = (const float*)d_in[8];
    const float* cls_w2 = (const float*)d_in[9];   // [256, 2]
    const float* cls_b2 = (const float*)d_in[10];
    float* out = (float*)d_out;

    // workspace layout (floats)
    float* f = (float*)d_ws;
    float* Xcat = f;                                   // 4096*2048
    float* Xn   = Xcat + (size_t)NN * DX;              // 4096*2048
    float* Xn3  = Xn + (size_t)NN * DX;                // 4096*8
    float* L    = Xn3 + (size_t)NN * METAD;            // 4096*4096  D1 -> adj -> Lhat
    float* D2   = L + (size_t)NN * NN;                 // 4096*4096  (reused as T1|T2)
    float* T1   = D2;                                  // 4096*2048
    float* T2   = D2 + (size_t)NN * DX;                // 4096*2048
    float* H0   = D2 + (size_t)NN * NN;                // 4096*256
    float* U1   = H0 + (size_t)NN * HG;
    float* U2   = U1 + (size_t)NN * HG;
    float* H1   = U2 + (size_t)NN * HG;
    float* Zb   = H1 + (size_t)NN * HG;
    float* deg  = Zb + (size_t)NN * HG;                // 4096
    float* dinv = deg + NN;                            // 4096
    float* musum = dinv + NN;                          // 256
    float* sqsum = musum + HG;                         // 256
    double* dsum = (double*)(sqsum + HG);              // 2 doubles (8B aligned)

    k_zero_doubles<<<1, 32, 0, stream>>>(dsum, 2);
    k_rownorm_cat<<<NN, 256, 0, stream>>>(x1, x2, Xcat, Xn);
    k_rownorm_meta<<<NN / 256, 256, 0, stream>>>(x3, Xn3);

    // distance matrices via WMMA syrk-style NT GEMM (16x64 per wave)
    dim3 gdist(NN / (16 * NT), NN / 128);
    k_dist_nt<<<gdist, 256, 0, stream>>>(Xn, DX, DX, L);
    k_dist_nt<<<gdist, 256, 0, stream>>>(Xn3, METAD, METAD, D2);

    const long long nn2 = (long long)NN * NN;
    k_sum_f64<<<2048, 256, 0, stream>>>(L, nn2, dsum + 0);
    k_sum_f64<<<2048, 256, 0, stream>>>(D2, nn2, dsum + 1);

    k_adj_deg<<<NN, 256, 0, stream>>>(L, D2, dsum, deg);
    k_dinv<<<NN / 256, 256, 0, stream>>>(deg, dinv);
    k_lhat<<<4096, 256, 0, stream>>>(L, dinv);

    // Cheb layer 0:  T1 = L@X ; T2 = 2*L@T1 - X ; H0 = relu(X@W00 + T1@W01 + T2@W02)
    k_gemm_nn<<<dim3(DX / (16 * NT), NN / 128), 256, 0, stream>>>(L, Xcat, nullptr, T1, DX, NN, 1.0f, 0.0f);
    k_gemm_nn<<<dim3(DX / (16 * NT), NN / 128), 256, 0, stream>>>(L, T1, Xcat, T2, DX, NN, 2.0f, -1.0f);
    k_gemm3_relu<<<dim3(HG / (16 * NT), NN / 128), 256, 0, stream>>>(
        Xcat, T1, T2,
        gc_w0, gc_w0 + (size_t)DX * HG, gc_w0 + 2 * (size_t)DX * HG,
        H0, HG, DX);

    // Cheb layer 1
    k_gemm_nn<<<dim3(HG / (16 * NT), NN / 128), 256, 0, stream>>>(L, H0, nullptr, U1, HG, NN, 1.0f, 0.0f);
    k_gemm_nn<<<dim3(HG / (16 * NT), NN / 128), 256, 0, stream>>>(L, U1, H0, U2, HG, NN, 2.0f, -1.0f);
    k_gemm3_relu<<<dim3(HG / (16 * NT), NN / 128), 256, 0, stream>>>(
        H0, U1, U2,
        gc_w1, gc_w1 + (size_t)HG * HG, gc_w1 + 2 * (size_t)HG * HG,
        H1, HG, HG);

    // classifier
    k_gemm_bias_relu<<<dim3(HG / (16 * NT), NN / 128), 256, 0, stream>>>(H1, cls_w1, cls_b1, Zb, HG, HG);
    k_colstats<<<HG, 256, 0, stream>>>(Zb, musum, sqsum, NN, HG);
    k_final<<<NN / 256, 256, 0, stream>>>(Zb, musum, sqsum, bn_g, bn_b, cls_w2, cls_b2, out, NN);
}